// MatrixReLU_17652315586744
// MI455X (gfx1250) — compile-verified
//
#include <hip/hip_runtime.h>
#include <hip/hip_bf16.h>

typedef float v2f __attribute__((ext_vector_type(2)));
typedef float v8f __attribute__((ext_vector_type(8)));

#define EPS       1e-4f
#define NS_ITERS  30
#define LDS       68   // uniform padded stride: conflict-free A reads + C stores, 16B rows

// Two 16x16 tiles (same tile column tn, rows tm0 and tm0+1) of A[64x64]*B[64x64].
// B fragments are shared between the two v_wmma issues per k-step.
// A per lane: r=lane%16, h=lane/16 -> {A[row][4kc+2h], A[row][4kc+2h+1]}  (one ds_load_b64)
// B per lane: c=lane%16            -> {B[4kc+2h][c],  B[4kc+2h+1][c]}
__device__ __forceinline__ void wmma_colpair64(const float* __restrict__ A, int lda,
                                               const float* __restrict__ B, int ldb,
                                               int tm0, int tn, int lane,
                                               v8f& acc0, v8f& acc1)
{
    const int r = lane & 15;
    const int h = lane >> 4;
    const float* Arow0 = A + (tm0 * 16 + r) * lda;
    const float* Arow1 = Arow0 + 16 * lda;
    const float* Bcol  = B + tn * 16 + r;
#pragma unroll
    for (int kc = 0; kc < 16; ++kc) {
        const int k = kc * 4 + 2 * h;                 // k even -> 8B-aligned v2f loads
        const v2f a0 = *(const v2f*)(Arow0 + k);
        const v2f a1 = *(const v2f*)(Arow1 + k);
        v2f b;
        b.x = Bcol[k * ldb];
        b.y = Bcol[(k + 1) * ldb];
        acc0 = __builtin_amdgcn_wmma_f32_16x16x4_f32(false, a0, false, b,
                                                     (short)0, acc0, false, false);
        acc1 = __builtin_amdgcn_wmma_f32_16x16x4_f32(false, a1, false, b,
                                                     (short)0, acc1, false, false);
    }
}

// C/D layout: element v of lane -> row tm*16 + v + 8*h, col tn*16 + r
__device__ __forceinline__ void store_tile64(float* __restrict__ D,
                                             int tm, int tn, int lane, v8f acc)
{
    const int r = lane & 15;
    const int h = lane >> 4;
#pragma unroll
    for (int v = 0; v < 8; ++v)
        D[(tm * 16 + v + 8 * h) * LDS + tn * 16 + r] = acc[v];
}

// out = U max(S,eps) U^T == 0.5*(X - eps*I)*(I + sign(X - eps*I)) + eps*I
__global__ __launch_bounds__(256) void matrix_relu_spectral(const float* __restrict__ X,
                                                            float* __restrict__ Out)
{
    __shared__ __align__(16) float sS0[64 * LDS];
    __shared__ __align__(16) float sS1[64 * LDS];
    __shared__ __align__(16) float sT [64 * LDS];

    const int tid  = threadIdx.x;
    const int lane = tid & 31;
    const int wave = tid >> 5;
    // tile assignment: column-paired -> shared B fragments
    const int tn  = wave & 3;
    const int tm0 = (wave >> 2) * 2;
    const long long base = (long long)blockIdx.x * 4096;
    const float* __restrict__ Xm = X + base;
    float* __restrict__ Om = Out + base;

    // ---- Load B = X - eps*I into sS0 (unscaled); accumulate Frobenius^2 ----
    float ss = 0.f;
#pragma unroll
    for (int j = 0; j < 4; ++j) {
        const int idx = tid + 256 * j;      // float4 index in [0,1024)
        const int r = idx >> 4;
        const int c = (idx & 15) << 2;
        float4 v = ((const float4*)Xm)[idx];
        if (r == c + 0) v.x -= EPS;
        if (r == c + 1) v.y -= EPS;
        if (r == c + 2) v.z -= EPS;
        if (r == c + 3) v.w -= EPS;
        ss += v.x * v.x + v.y * v.y + v.z * v.z + v.w * v.w;
        *(float4*)(&sS0[r * LDS + c]) = v;  // 272B row stride -> 16B aligned
    }
#pragma unroll
    for (int o = 16; o > 0; o >>= 1) ss += __shfl_down(ss, o, 32);
    if (lane == 0) sT[wave] = ss;
    __syncthreads();
    if (tid == 0) {
        float t = 0.f;
#pragma unroll
        for (int w = 0; w < 8; ++w) t += sT[w];
        sT[8] = t;
    }
    __syncthreads();
    const float fro2  = sT[8];
    const float scale = (fro2 > 1e-30f) ? rsqrtf(fro2) : 0.f;

    // ---- S0 = B / ||B||_F ----
#pragma unroll
    for (int j = 0; j < 16; ++j) {
        const int e = tid + 256 * j;
        sS0[(e >> 6) * LDS + (e & 63)] *= scale;
    }
    __syncthreads();

    // ---- Newton-Schulz: S' = 1.5*S - 0.5*S*(S*S), transform fused in registers ----
    float* S    = sS0;
    float* Snew = sS1;
    const int r = lane & 15;
    const int h = lane >> 4;
    const int col = tn * 16 + r;
    for (int it = 0; it < NS_ITERS; ++it) {
        // T = S * S
        v8f t0 = {}, t1 = {};
        wmma_colpair64(S, LDS, S, LDS, tm0, tn, lane, t0, t1);
        store_tile64(sT, tm0,     tn, lane, t0);
        store_tile64(sT, tm0 + 1, tn, lane, t1);
        __syncthreads();
        // M = S * T;  Snew = 1.5*S - 0.5*M  (no elementwise pass over T)
        v8f m0 = {}, m1 = {};
        wmma_colpair64(S, LDS, sT, LDS, tm0, tn, lane, m0, m1);
#pragma unroll
        for (int v = 0; v < 8; ++v) {
            const int row0 = tm0 * 16 + v + 8 * h;
            const int row1 = row0 + 16;
            Snew[row0 * LDS + col] = 1.5f * S[row0 * LDS + col] - 0.5f * m0[v];
            Snew[row1 * LDS + col] = 1.5f * S[row1 * LDS + col] - 0.5f * m1[v];
        }
        __syncthreads();
        float* tmp = S; S = Snew; Snew = tmp;
    }

    // ---- S <- I + S ----
    if (tid < 64) S[tid * LDS + tid] += 1.0f;
    __syncthreads();

    // ---- R = B * (I + S): A-fragments streamed from global X (L2-resident) ----
    {
        const float* __restrict__ Xrow0 = Xm + (tm0 * 16 + r) * 64;
        const float* __restrict__ Xrow1 = Xrow0 + 16 * 64;
        const int row0g = tm0 * 16 + r, row1g = row0g + 16;
        const float* __restrict__ Scol = S + col;
        v8f acc0 = {}, acc1 = {};
#pragma unroll
        for (int kc = 0; kc < 16; ++kc) {
            const int k = kc * 4 + 2 * h;
            v2f a0 = *(const v2f*)(Xrow0 + k);
            v2f a1 = *(const v2f*)(Xrow1 + k);
            a0.x -= (row0g == k)     ? EPS : 0.f;
            a0.y -= (row0g == k + 1) ? EPS : 0.f;
            a1.x -= (row1g == k)     ? EPS : 0.f;
            a1.y -= (row1g == k + 1) ? EPS : 0.f;
            v2f b;
            b.x = Scol[k * LDS];
            b.y = Scol[(k + 1) * LDS];
            acc0 = __builtin_amdgcn_wmma_f32_16x16x4_f32(false, a0, false, b,
                                                         (short)0, acc0, false, false);
            acc1 = __builtin_amdgcn_wmma_f32_16x16x4_f32(false, a1, false, b,
                                                         (short)0, acc1, false, false);
        }
        // stage Out = 0.5*R + eps*I into sT for coalesced writeback
#pragma unroll
        for (int v = 0; v < 8; ++v) {
            const int R0 = tm0 * 16 + v + 8 * h;
            const int R1 = R0 + 16;
            float o0 = 0.5f * acc0[v];
            float o1 = 0.5f * acc1[v];
            if (R0 == col) o0 += EPS;
            if (R1 == col) o1 += EPS;
            sT[R0 * LDS + col] = o0;
            sT[R1 * LDS + col] = o1;
        }
    }
    __syncthreads();

    // ---- coalesced float4 writeback ----
#pragma unroll
    for (int j = 0; j < 4; ++j) {
        const int idx = tid + 256 * j;
        const int r2 = idx >> 4;
        const int c2 = (idx & 15) << 2;
        const float4 v = *(const float4*)(&sT[r2 * LDS + c2]);
        ((float4*)Om)[idx] = v;
    }
}

extern "C" void kernel_launch(void* const* d_in, const int* in_sizes, int n_in,
                              void* d_out, int out_size, void* d_ws, size_t ws_size,
                              hipStream_t stream) {
    (void)n_in; (void)out_size; (void)d_ws; (void)ws_size;
    const float* x = (const float*)d_in[0];
    float* out = (float*)d_out;
    const int nmat = in_sizes[0] / 4096;   // 8192 matrices of 64*64
    matrix_relu_spectral<<<dim3(nmat), dim3(256), 0, stream>>>(x, out);
}